// RSSM_77103252898298
// MI455X (gfx1250) — compile-verified
//
#include <hip/hip_runtime.h>
#include <hip/hip_bf16.h>
#include <math.h>

#define DIMX 100
#define DIMZ 100
#define DIMH 10
#define DIMG 10
#define DIMDX 20
#define BB 256
#define TT 512
#define G3 30
#define H3 30

typedef __attribute__((ext_vector_type(2))) float v2f;
typedef __attribute__((ext_vector_type(8))) float v8f;

// D = A(16x4, f32) * B(4x16, f32) + C(16x16, f32), full precision.
// 8-arg modifier form: (neg_a, A, neg_b, B, c_mod, C, reuse_a, reuse_b)
__device__ __forceinline__ v8f wmma4(v2f a, v2f b, v8f c) {
  return __builtin_amdgcn_wmma_f32_16x16x4_f32(false, a, false, b, (short)0, c,
                                               false, false);
}

__device__ __forceinline__ float sigmoidf_(float x) {
  return 1.0f / (1.0f + __expf(-x));
}
__device__ __forceinline__ float softplusf_(float x) {
  return (x > 20.0f) ? x : log1pf(__expf(x));
}

// ---------------------------------------------------------------------------
// Kernel A: gi_inf[r][0:32] = x[r][:] @ inf_Wih.T + bih   for r in [0, B*T)
// Fully parallel WMMA GEMM: M=B*T (16-row tiles), N=30->32, K=100 (25 steps).
// A-tiles (16x100 f32, contiguous 6400B of x per wave) are staged to LDS with
// GLOBAL_LOAD_ASYNC_TO_LDS_B128 (ASYNCcnt), then read bank-conflict-free
// (row stride 100 dwords -> 16 distinct banks across the 16 rows).
// ---------------------------------------------------------------------------
__global__ __launch_bounds__(256) void k_input_proj(
    const float* __restrict__ x, const float* __restrict__ Wih,
    const float* __restrict__ bih, float* __restrict__ gi) {
  __shared__ float Bw[DIMX][32];  // Bw[k][n] = Wih[n][k]
  __shared__ __attribute__((aligned(16))) float xtile[8][16 * DIMX];
  for (int i = threadIdx.x; i < DIMX * 32; i += 256) {
    int k = i >> 5, n = i & 31;
    Bw[k][n] = (n < G3) ? Wih[n * DIMX + k] : 0.0f;
  }
  __syncthreads();

  int lane = threadIdx.x & 31;
  int wave = threadIdx.x >> 5;
  size_t r0 = ((size_t)blockIdx.x * 8 + wave) * 16;
  int row = lane & 15;
  int hi = lane >> 4;  // half-wave selects K pair {0,1} vs {2,3}
  int col = lane & 15;

  // --- async DMA: 16 rows x 100 f32 = 400 contiguous float4 -> LDS ---
  {
    const float* gsrc = x + r0 * DIMX;  // 16B-aligned (r0 multiple of 16)
    unsigned ldsbase = (unsigned)(uintptr_t)(&xtile[wave][0]);
    for (int q = lane; q < (16 * DIMX) / 4; q += 32) {
      unsigned ldsa = ldsbase + (unsigned)q * 16u;
      unsigned long long ga = (unsigned long long)(uintptr_t)(gsrc + q * 4);
      asm volatile("global_load_async_to_lds_b128 %0, %1, off"
                   :
                   : "v"(ldsa), "v"(ga)
                   : "memory");
    }
    asm volatile("s_wait_asynccnt 0" ::: "memory");
  }

  // Hoist all 25 A fragments into VGPRs (reused by both N tiles).
  v2f afr[25];
  const float* xrow = &xtile[wave][row * DIMX];
#pragma unroll
  for (int kk = 0; kk < 25; ++kk) {
    int ka = kk * 4 + hi * 2;
    afr[kk].x = xrow[ka];
    afr[kk].y = xrow[ka + 1];
  }

  float* gout = gi + r0 * 32;
  for (int nt = 0; nt < 2; ++nt) {
    int n0 = nt * 16;
    float bias = (n0 + col < G3) ? bih[n0 + col] : 0.0f;
    v8f acc;
#pragma unroll
    for (int i = 0; i < 8; ++i) acc[i] = bias;
#pragma unroll
    for (int kk = 0; kk < 25; ++kk) {
      int ka = kk * 4 + hi * 2;
      v2f b;
      b.x = Bw[ka][n0 + col];
      b.y = Bw[ka + 1][n0 + col];
      acc = wmma4(afr[kk], b, acc);
    }
#pragma unroll
    for (int i = 0; i < 8; ++i) {
      int m = i + hi * 8;
      gout[(size_t)m * 32 + n0 + col] = acc[i];
    }
  }
}

// ---------------------------------------------------------------------------
// Kernel B: reverse-time inference GRU. One wave per 16-batch tile.
// Per step: gh = h @ Whh.T + bhh via 6 f32 WMMAs, gates in VALU, h in LDS.
// ---------------------------------------------------------------------------
__global__ __launch_bounds__(32) void k_inf_scan(
    const float* __restrict__ gi, const float* __restrict__ Whh,
    const float* __restrict__ bhh, const float* __restrict__ g0,
    float* __restrict__ gseq) {
  __shared__ float Bw[12][32];    // Bw[k][n] = Whh[n][k], zero padded
  __shared__ float hbuf[16][12];  // recurrent state, K-padded
  __shared__ float ghb[16][32];   // D-fragment spill for gate math
  int lane = threadIdx.x;
  for (int i = lane; i < 12 * 32; i += 32) {
    int k = i >> 5, n = i & 31;
    Bw[k][n] = (k < DIMG && n < G3) ? Whh[n * DIMG + k] : 0.0f;
  }
  for (int i = lane; i < 16 * 12; i += 32) {
    int m = i / 12, j = i % 12;
    hbuf[m][j] = (j < DIMG) ? g0[j] : 0.0f;
  }
  __syncthreads();

  int b0 = blockIdx.x * 16;
  int row = lane & 15, hi = lane >> 4, col = lane & 15;
  int jb = hi * 5;  // lanes 0-15 handle gate cols 0-4, lanes 16-31 cols 5-9

  for (int tt = 0; tt < TT; ++tt) {
    int t = TT - 1 - tt;
    for (int nt = 0; nt < 2; ++nt) {
      int n0 = nt * 16;
      float bias = (n0 + col < G3) ? bhh[n0 + col] : 0.0f;
      v8f acc;
#pragma unroll
      for (int i = 0; i < 8; ++i) acc[i] = bias;
#pragma unroll
      for (int kk = 0; kk < 12; kk += 4) {
        int ka = kk + hi * 2;
        v2f a, b;
        a.x = hbuf[row][ka];
        a.y = hbuf[row][ka + 1];
        b.x = Bw[ka][n0 + col];
        b.y = Bw[ka + 1][n0 + col];
        acc = wmma4(a, b, acc);
      }
#pragma unroll
      for (int i = 0; i < 8; ++i) ghb[i + hi * 8][n0 + col] = acc[i];
    }
    __syncthreads();

    const float* girow = gi + ((size_t)(b0 + row) * TT + t) * 32;
    float* grow = gseq + ((size_t)(b0 + row) * TT + t) * 16;
    if (t > 0) __builtin_prefetch(girow - 32, 0, 1);  // next step's gi
#pragma unroll
    for (int jj = 0; jj < 5; ++jj) {
      int j = jb + jj;
      float rr = sigmoidf_(girow[j] + ghb[row][j]);
      float zz = sigmoidf_(girow[10 + j] + ghb[row][10 + j]);
      float nn = tanhf(girow[20 + j] + rr * ghb[row][20 + j]);
      float hv = (1.0f - zz) * nn + zz * hbuf[row][j];
      grow[j] = hv;
      hbuf[row][j] = hv;
    }
    __syncthreads();
  }
}

// ---------------------------------------------------------------------------
// Kernel C0: fold gen_Wih(30x100) @ cmb_Wloc(100x10) -> M(30x10) and
// b' = gen_Wih @ cmb_bloc + gen_bih.  Removes the 16x100x30 matmul from the
// recurrent critical path.
// ---------------------------------------------------------------------------
__global__ __launch_bounds__(416) void k_fold(
    const float* __restrict__ genWih, const float* __restrict__ Wloc,
    const float* __restrict__ bloc, const float* __restrict__ genBih,
    float* __restrict__ Mfold, float* __restrict__ bfold) {
  int i = threadIdx.x;
  if (i < 32 * 12) {
    int r = i / 12, c = i % 12;
    float s = 0.0f;
    if (r < H3 && c < DIMG)
      for (int k = 0; k < DIMZ; ++k) s += genWih[r * DIMZ + k] * Wloc[k * DIMG + c];
    Mfold[i] = s;  // row-major [32][12], zero padded
  } else if (i < 32 * 12 + 32) {
    int r = i - 32 * 12;
    float s = 0.0f;
    if (r < H3) {
      for (int k = 0; k < DIMZ; ++k) s += genWih[r * DIMZ + k] * bloc[k];
      s += genBih[r];
    }
    bfold[r] = s;
  }
}

// ---------------------------------------------------------------------------
// Kernel C: forward generative GRU scan with folded input matrix.
// Records h_comb[t] and h_new[t] (10 floats each) for the parallel emitter.
// ---------------------------------------------------------------------------
__global__ __launch_bounds__(32) void k_gen_scan(
    const float* __restrict__ gseq, const float* __restrict__ Wh2g,
    const float* __restrict__ bh2g, const float* __restrict__ Mfold,
    const float* __restrict__ bfold, const float* __restrict__ Whh,
    const float* __restrict__ bhh, const float* __restrict__ h0,
    float* __restrict__ hcomb_seq, float* __restrict__ hnew_seq) {
  __shared__ float B2g[12][16];   // Wh2g^T padded
  __shared__ float BM[12][32];    // Mfold^T padded
  __shared__ float Bwh[12][32];   // gen_Whh^T padded
  __shared__ float hbuf[16][12];
  __shared__ float hcb[16][12];
  __shared__ float tmp[16][16];
  __shared__ float gib[16][32];
  __shared__ float ghb[16][32];
  int lane = threadIdx.x;
  for (int i = lane; i < 12 * 16; i += 32) {
    int k = i >> 4, n = i & 15;
    B2g[k][n] = (k < DIMH && n < DIMG) ? Wh2g[n * DIMH + k] : 0.0f;
  }
  for (int i = lane; i < 12 * 32; i += 32) {
    int k = i >> 5, n = i & 31;
    BM[k][n] = Mfold[n * 12 + k];  // already zero padded by k_fold
    Bwh[k][n] = (k < DIMH && n < H3) ? Whh[n * DIMH + k] : 0.0f;
  }
  for (int i = lane; i < 16 * 12; i += 32) {
    int m = i / 12, j = i % 12;
    float v = (j < DIMH) ? h0[j] : 0.0f;
    hbuf[m][j] = v;
    hcb[m][j] = 0.0f;
  }
  __syncthreads();

  int b0 = blockIdx.x * 16;
  int row = lane & 15, hi = lane >> 4, col = lane & 15;
  int jb = hi * 5;

  for (int t = 0; t < TT; ++t) {
    // --- h_comb pre-activation: h @ Wh2g^T + bh2g (one 16-col tile) ---
    {
      float bias = (col < DIMG) ? bh2g[col] : 0.0f;
      v8f acc;
#pragma unroll
      for (int i = 0; i < 8; ++i) acc[i] = bias;
#pragma unroll
      for (int kk = 0; kk < 12; kk += 4) {
        int ka = kk + hi * 2;
        v2f a, b;
        a.x = hbuf[row][ka];
        a.y = hbuf[row][ka + 1];
        b.x = B2g[ka][col];
        b.y = B2g[ka + 1][col];
        acc = wmma4(a, b, acc);
      }
#pragma unroll
      for (int i = 0; i < 8; ++i) tmp[i + hi * 8][col] = acc[i];
    }
    __syncthreads();
    {
      const float* grow = gseq + ((size_t)(b0 + row) * TT + t) * 16;
      float* hcrow = hcomb_seq + ((size_t)(b0 + row) * TT + t) * 16;
      if (t + 1 < TT) __builtin_prefetch(grow + 16, 0, 1);
#pragma unroll
      for (int jj = 0; jj < 5; ++jj) {
        int j = jb + jj;
        float hc = 0.5f * (tanhf(tmp[row][j]) + grow[j]);
        hcb[row][j] = hc;
        hcrow[j] = hc;
      }
    }
    __syncthreads();
    // --- gi = h_comb @ M^T + b' ; gh = h @ Whh^T + bhh (2 tiles each) ---
    for (int nt = 0; nt < 2; ++nt) {
      int n0 = nt * 16;
      float biasg = bfold[n0 + col];
      float biash = (n0 + col < H3) ? bhh[n0 + col] : 0.0f;
      v8f ai, ah;
#pragma unroll
      for (int i = 0; i < 8; ++i) { ai[i] = biasg; ah[i] = biash; }
#pragma unroll
      for (int kk = 0; kk < 12; kk += 4) {
        int ka = kk + hi * 2;
        v2f a1, a2, b1, b2;
        a1.x = hcb[row][ka];  a1.y = hcb[row][ka + 1];
        a2.x = hbuf[row][ka]; a2.y = hbuf[row][ka + 1];
        b1.x = BM[ka][n0 + col];  b1.y = BM[ka + 1][n0 + col];
        b2.x = Bwh[ka][n0 + col]; b2.y = Bwh[ka + 1][n0 + col];
        ai = wmma4(a1, b1, ai);
        ah = wmma4(a2, b2, ah);
      }
#pragma unroll
      for (int i = 0; i < 8; ++i) {
        gib[i + hi * 8][n0 + col] = ai[i];
        ghb[i + hi * 8][n0 + col] = ah[i];
      }
    }
    __syncthreads();
    {
      float* hnrow = hnew_seq + ((size_t)(b0 + row) * TT + t) * 16;
#pragma unroll
      for (int jj = 0; jj < 5; ++jj) {
        int j = jb + jj;
        float rr = sigmoidf_(gib[row][j] + ghb[row][j]);
        float zz = sigmoidf_(gib[row][10 + j] + ghb[row][10 + j]);
        float nn = tanhf(gib[row][20 + j] + rr * ghb[row][20 + j]);
        float hv = (1.0f - zz) * nn + zz * hbuf[row][j];
        hnrow[j] = hv;
        hbuf[row][j] = hv;
      }
    }
    __syncthreads();
  }
}

// ---------------------------------------------------------------------------
// Kernel D: fully parallel output emitter. Per 16-row tile: z_loc/z_scale
// from h_comb (K=12, N=100), hid=relu(h_new@Whid^T) (K=12, N=20), then
// x_loc/x_scale (K=20, N=100). ~118 f32 WMMAs per wave; HBM-write bound.
// ---------------------------------------------------------------------------
__global__ __launch_bounds__(256) void k_emit(
    const float* __restrict__ hcomb_seq, const float* __restrict__ hnew_seq,
    const float* __restrict__ Wloc, const float* __restrict__ bloc,
    const float* __restrict__ Wscale, const float* __restrict__ bscale,
    const float* __restrict__ Whid, const float* __restrict__ bhid,
    const float* __restrict__ Wxloc, const float* __restrict__ bxloc,
    const float* __restrict__ Wxscale, const float* __restrict__ bxscale,
    float* __restrict__ out) {
  __shared__ float BL[12][112];
  __shared__ float BS[12][112];
  __shared__ float BH[12][32];
  __shared__ float BXL[20][112];
  __shared__ float BXS[20][112];
  __shared__ float hidb[8][16][20];
  for (int i = threadIdx.x; i < 12 * 112; i += 256) {
    int k = i / 112, n = i % 112;
    bool ok = (k < DIMG && n < DIMZ);
    BL[k][n] = ok ? Wloc[n * DIMG + k] : 0.0f;
    BS[k][n] = ok ? Wscale[n * DIMG + k] : 0.0f;
  }
  for (int i = threadIdx.x; i < 12 * 32; i += 256) {
    int k = i >> 5, n = i & 31;
    BH[k][n] = (k < DIMH && n < DIMDX) ? Whid[n * DIMH + k] : 0.0f;
  }
  for (int i = threadIdx.x; i < 20 * 112; i += 256) {
    int k = i / 112, n = i % 112;
    BXL[k][n] = (n < DIMX) ? Wxloc[n * DIMDX + k] : 0.0f;
    BXS[k][n] = (n < DIMX) ? Wxscale[n * DIMDX + k] : 0.0f;
  }
  __syncthreads();

  int lane = threadIdx.x & 31, wave = threadIdx.x >> 5;
  size_t r0 = ((size_t)blockIdx.x * 8 + wave) * 16;
  int row = lane & 15, hi = lane >> 4, col = lane & 15;

  // A fragments for h_comb and h_new (K padded 10->12)
  v2f ac[3], an[3];
  const float* hcrow = hcomb_seq + (r0 + row) * 16;
  const float* hnrow = hnew_seq + (r0 + row) * 16;
#pragma unroll
  for (int kk = 0; kk < 3; ++kk) {
    int ka = kk * 4 + hi * 2;
    ac[kk].x = (ka < DIMG) ? hcrow[ka] : 0.0f;
    ac[kk].y = (ka + 1 < DIMG) ? hcrow[ka + 1] : 0.0f;
    an[kk].x = (ka < DIMH) ? hnrow[ka] : 0.0f;
    an[kk].y = (ka + 1 < DIMH) ? hnrow[ka + 1] : 0.0f;
  }

  // --- z_loc (seg 0) and z_scale (seg 100) ---
  for (int nt = 0; nt < 7; ++nt) {
    int n0 = nt * 16, n = n0 + col;
    bool ok = (n < DIMZ);
    v8f al, as;
    float bl = ok ? bloc[n] : 0.0f;
    float bs = ok ? bscale[n] : 0.0f;
#pragma unroll
    for (int i = 0; i < 8; ++i) { al[i] = bl; as[i] = bs; }
#pragma unroll
    for (int kk = 0; kk < 3; ++kk) {
      int ka = kk * 4 + hi * 2;
      v2f b1, b2;
      b1.x = BL[ka][n]; b1.y = BL[ka + 1][n];
      b2.x = BS[ka][n]; b2.y = BS[ka + 1][n];
      al = wmma4(ac[kk], b1, al);
      as = wmma4(ac[kk], b2, as);
    }
    if (ok) {
#pragma unroll
      for (int i = 0; i < 8; ++i) {
        size_t r = r0 + i + hi * 8;
        out[r * 400 + n] = al[i];
        out[r * 400 + 100 + n] = softplusf_(as[i]);
      }
    }
  }

  // --- hid = relu(h_new @ Whid^T + bhid), spilled to LDS for re-fragmenting
  for (int nt = 0; nt < 2; ++nt) {
    int n0 = nt * 16, n = n0 + col;
    float bh = (n < DIMDX) ? bhid[n] : 0.0f;
    v8f ah;
#pragma unroll
    for (int i = 0; i < 8; ++i) ah[i] = bh;
#pragma unroll
    for (int kk = 0; kk < 3; ++kk) {
      int ka = kk * 4 + hi * 2;
      v2f b;
      b.x = BH[ka][n]; b.y = BH[ka + 1][n];
      ah = wmma4(an[kk], b, ah);
    }
    if (n < DIMDX) {
#pragma unroll
      for (int i = 0; i < 8; ++i) hidb[wave][i + hi * 8][n] = fmaxf(ah[i], 0.0f);
    }
  }
  __syncthreads();

  v2f ahid[5];
#pragma unroll
  for (int kk = 0; kk < 5; ++kk) {
    int ka = kk * 4 + hi * 2;
    ahid[kk].x = hidb[wave][row][ka];
    ahid[kk].y = hidb[wave][row][ka + 1];
  }

  // --- x_loc (seg 200) and x_scale (seg 300) ---
  for (int nt = 0; nt < 7; ++nt) {
    int n0 = nt * 16, n = n0 + col;
    bool ok = (n < DIMX);
    v8f al, as;
    float bl = ok ? bxloc[n] : 0.0f;
    float bs = ok ? bxscale[n] : 0.0f;
#pragma unroll
    for (int i = 0; i < 8; ++i) { al[i] = bl; as[i] = bs; }
#pragma unroll
    for (int kk = 0; kk < 5; ++kk) {
      int ka = kk * 4 + hi * 2;
      v2f b1, b2;
      b1.x = BXL[ka][n]; b1.y = BXL[ka + 1][n];
      b2.x = BXS[ka][n]; b2.y = BXS[ka + 1][n];
      al = wmma4(ahid[kk], b1, al);
      as = wmma4(ahid[kk], b2, as);
    }
    if (ok) {
#pragma unroll
      for (int i = 0; i < 8; ++i) {
        size_t r = r0 + i + hi * 8;
        out[r * 400 + 200 + n] = al[i];
        out[r * 400 + 300 + n] = softplusf_(as[i]);
      }
    }
  }
}

extern "C" void kernel_launch(void* const* d_in, const int* in_sizes, int n_in,
                              void* d_out, int out_size, void* d_ws,
                              size_t ws_size, hipStream_t stream) {
  (void)in_sizes; (void)n_in; (void)out_size; (void)ws_size;
  const float* x          = (const float*)d_in[0];
  const float* inf_Wih    = (const float*)d_in[1];
  const float* inf_Whh    = (const float*)d_in[2];
  const float* inf_bih    = (const float*)d_in[3];
  const float* inf_bhh    = (const float*)d_in[4];
  const float* gen_Wih    = (const float*)d_in[5];
  const float* gen_Whh    = (const float*)d_in[6];
  const float* gen_bih    = (const float*)d_in[7];
  const float* gen_bhh    = (const float*)d_in[8];
  const float* cmb_Wh2g   = (const float*)d_in[9];
  const float* cmb_bh2g   = (const float*)d_in[10];
  const float* cmb_Wloc   = (const float*)d_in[11];
  const float* cmb_bloc   = (const float*)d_in[12];
  const float* cmb_Wscale = (const float*)d_in[13];
  const float* cmb_bscale = (const float*)d_in[14];
  const float* dx_Whid    = (const float*)d_in[15];
  const float* dx_bhid    = (const float*)d_in[16];
  const float* dx_Wloc    = (const float*)d_in[17];
  const float* dx_bloc    = (const float*)d_in[18];
  const float* dx_Wscale  = (const float*)d_in[19];
  const float* dx_bscale  = (const float*)d_in[20];
  const float* h_0        = (const float*)d_in[21];
  const float* g_0        = (const float*)d_in[22];

  float* ws = (float*)d_ws;
  float* gi_inf = ws;                                  // B*T*32
  float* g_seq  = gi_inf + (size_t)BB * TT * 32;       // B*T*16
  float* hcomb  = g_seq  + (size_t)BB * TT * 16;       // B*T*16
  float* hnew   = hcomb  + (size_t)BB * TT * 16;       // B*T*16
  float* Mfold  = hnew   + (size_t)BB * TT * 16;       // 32*12
  float* bfold  = Mfold + 32 * 12;                     // 32

  k_fold<<<1, 416, 0, stream>>>(gen_Wih, cmb_Wloc, cmb_bloc, gen_bih, Mfold,
                                bfold);
  k_input_proj<<<1024, 256, 0, stream>>>(x, inf_Wih, inf_bih, gi_inf);
  k_inf_scan<<<16, 32, 0, stream>>>(gi_inf, inf_Whh, inf_bhh, g_0, g_seq);
  k_gen_scan<<<16, 32, 0, stream>>>(g_seq, cmb_Wh2g, cmb_bh2g, Mfold, bfold,
                                    gen_Whh, gen_bhh, h_0, hcomb, hnew);
  k_emit<<<1024, 256, 0, stream>>>(hcomb, hnew, cmb_Wloc, cmb_bloc, cmb_Wscale,
                                   cmb_bscale, dx_Whid, dx_bhid, dx_Wloc,
                                   dx_bloc, dx_Wscale, dx_bscale,
                                   (float*)d_out);
}